// CrossAttention_26096221290564
// MI455X (gfx1250) — compile-verified
//
#include <hip/hip_runtime.h>
#include <math.h>

// ---------------------------------------------------------------------------
// Fused cross-attention for MI455X (gfx1250, wave32, WMMA).
//   B=32, S=1024, d_model=48, H=3, d_k=16, fp32 everywhere.
// HBM-write bound by A (402 MB) -> single pass, NT stores for A,
// fp32 WMMA 16x16x4 for all matmuls (compute is ~7 GFLOP, irrelevant).
// ---------------------------------------------------------------------------

#define BATCH     32
#define SLEN      1024
#define DMODEL    48
#define NHEADS    3
#define DK        16
#define NROWS     (BATCH * SLEN)          // 32768
#define QKV_ELEMS (BATCH * NHEADS * SLEN * DK)   // 1,572,864 floats per tensor
#define H_ELEMS   (NROWS * DMODEL)               // 1,572,864 floats
#define PAD_STRIDE 1028                   // 1024 + 4 -> conflict-free acc stores

typedef __attribute__((ext_vector_type(2))) float v2f;
typedef __attribute__((ext_vector_type(8))) float v8f;

// D = A(16x4 f32) * B(4x16 f32) + C(16x16 f32)  -- V_WMMA_F32_16X16X4_F32
__device__ __forceinline__ v8f wmma4(v2f a, v2f b, v8f c) {
    return __builtin_amdgcn_wmma_f32_16x16x4_f32(
        /*neg_a=*/false, a, /*neg_b=*/false, b,
        /*c_mod=*/(short)0, c, /*reuse_a=*/false, /*reuse_b=*/false);
}

// ---------------------------------------------------------------------------
// Projection: out = X[32768,48] @ W[48,48]^T + bias.
//   One wave per 16-row tile; 3 accumulators = 3 head-sized N tiles.
//   split_heads=1 -> out laid out [B, H, S, 16] (attention-friendly)
//   split_heads=0 -> out laid out [row, 48]     (final H output)
// ---------------------------------------------------------------------------
__global__ __launch_bounds__(32)
void ca_proj_kernel(const float* __restrict__ X, const float* __restrict__ W,
                    const float* __restrict__ bias, float* __restrict__ out,
                    int split_heads)
{
    const int tile = blockIdx.x;          // 0..2047
    const int lane = threadIdx.x;         // wave32
    const int nm   = lane & 15;           // N (and M for A-frag rows)
    const int hi   = lane >> 4;           // half-wave selector
    const int koff = hi * 2;              // K offset within 4-wide step
    const int row0 = tile * 16;

    v8f c[NHEADS] = {};
    const float* xrow = X + (size_t)(row0 + nm) * DMODEL;

    #pragma unroll
    for (int k0 = 0; k0 < DMODEL; k0 += 4) {
        // A frag: X[m][k0+koff .. +1], m = nm
        v2f a = *(const v2f*)(xrow + k0 + koff);
        #pragma unroll
        for (int h = 0; h < NHEADS; ++h) {
            // B frag: B[k][n] = W[n][k], n = h*16 + nm  (contiguous in W row)
            v2f bf = *(const v2f*)(W + (size_t)(h * DK + nm) * DMODEL + k0 + koff);
            c[h] = wmma4(a, bf, c[h]);
        }
    }

    const int b  = row0 / SLEN;
    const int s0 = row0 % SLEN;           // SLEN % 16 == 0 -> tile within one b
    #pragma unroll
    for (int h = 0; h < NHEADS; ++h) {
        const float bv = bias[h * DK + nm];
        #pragma unroll
        for (int i = 0; i < 8; ++i) {
            const int m = i + hi * 8;     // C layout: rows i / i+8 per half-wave
            const float val = c[h][i] + bv;
            if (split_heads) {
                out[(((size_t)b * NHEADS + h) * SLEN + s0 + m) * DK + nm] = val;
            } else {
                out[((size_t)row0 + m) * DMODEL + h * DK + nm] = val;
            }
        }
    }
}

// ---------------------------------------------------------------------------
// Fused attention: per (b, h, 16-row q-tile):
//   scores[16,1024] in LDS via WMMA, softmax in place (+NT store of A),
//   O = P @ V via WMMA, O tile -> Hcat workspace.
// ---------------------------------------------------------------------------
__global__ __launch_bounds__(32)
void ca_attn_kernel(const float* __restrict__ Q, const float* __restrict__ K,
                    const float* __restrict__ V, float* __restrict__ Hcat,
                    float* __restrict__ Aout)
{
    __shared__ float sc[16 * PAD_STRIDE];          // 65,792 B of 320 KB WGP LDS

    const int qt   = blockIdx.x;          // 0..63 q tiles
    const int h    = blockIdx.y;          // 0..2
    const int b    = blockIdx.z;          // 0..31
    const int bh   = b * NHEADS + h;
    const int lane = threadIdx.x;
    const int nm   = lane & 15;
    const int hi   = lane >> 4;
    const int koff = hi * 2;

    // --- Q tile -> 4 A-fragments, pre-scaled by 1/sqrt(dk) = 0.25 -----------
    const float* qrow = Q + ((size_t)bh * SLEN + qt * 16 + nm) * DK;
    v2f qa[4];
    #pragma unroll
    for (int j = 0; j < 4; ++j) {
        v2f t = *(const v2f*)(qrow + j * 4 + koff);
        qa[j] = t * 0.25f;
    }

    // --- scores = (Q/sqrt(dk)) @ K^T, streamed into LDS ---------------------
    const float* Kb = K + (size_t)bh * SLEN * DK;
    for (int kt = 0; kt < SLEN / 16; ++kt) {
        const float* krow = Kb + (size_t)(kt * 16 + nm) * DK;
        v8f c = {};
        #pragma unroll
        for (int j = 0; j < 4; ++j) {
            // B frag: B[k][n] = K[kt*16+n][k]  (contiguous pair in K row)
            v2f bf = *(const v2f*)(krow + j * 4 + koff);
            c = wmma4(qa[j], bf, c);
        }
        #pragma unroll
        for (int i = 0; i < 8; ++i)
            sc[(i + hi * 8) * PAD_STRIDE + kt * 16 + nm] = c[i];
    }
    __syncthreads();   // single-wave WG: lowers to S_NOP, DS in-order anyway

    // --- row softmax in LDS; emit A with non-temporal stores ----------------
    const size_t arow_base = ((size_t)bh * SLEN + (size_t)qt * 16) * SLEN;
    for (int r = 0; r < 16; ++r) {
        float* srow = sc + r * PAD_STRIDE;
        float mx = -3.0e38f;
        for (int c0 = lane; c0 < SLEN; c0 += 32) mx = fmaxf(mx, srow[c0]);
        #pragma unroll
        for (int off = 16; off > 0; off >>= 1)
            mx = fmaxf(mx, __shfl_xor(mx, off, 32));

        float sum = 0.0f;
        for (int c0 = lane; c0 < SLEN; c0 += 32) {
            float e = __expf(srow[c0] - mx);
            srow[c0] = e;
            sum += e;
        }
        #pragma unroll
        for (int off = 16; off > 0; off >>= 1)
            sum += __shfl_xor(sum, off, 32);

        const float inv = 1.0f / sum;
        float* arow = Aout + arow_base + (size_t)r * SLEN;
        for (int c0 = lane; c0 < SLEN; c0 += 32) {
            float p = srow[c0] * inv;
            srow[c0] = p;
            __builtin_nontemporal_store(p, arow + c0);   // A never re-read: skip L2 residency
        }
    }
    __syncthreads();

    // --- O = P @ V ----------------------------------------------------------
    const float* Vb = V + (size_t)bh * SLEN * DK;
    v8f acc = {};
    for (int kt = 0; kt < SLEN / 16; ++kt) {
        #pragma unroll
        for (int j = 0; j < 4; ++j) {
            const int kk = kt * 16 + j * 4 + koff;
            // A frag: P[m][kk], P[m][kk+1] from LDS (8B aligned)
            v2f pa = *(const v2f*)(sc + (size_t)nm * PAD_STRIDE + kk);
            // B frag: V[kk][n], V[kk+1][n]
            v2f vb;
            vb[0] = Vb[(size_t)kk * DK + nm];
            vb[1] = Vb[(size_t)(kk + 1) * DK + nm];
            acc = wmma4(pa, vb, acc);
        }
    }
    #pragma unroll
    for (int i = 0; i < 8; ++i) {
        const int srowi = qt * 16 + i + hi * 8;
        Hcat[((size_t)b * SLEN + srowi) * DMODEL + h * DK + nm] = acc[i];
    }
}

// ---------------------------------------------------------------------------
// Launch: 3 projections -> fused attention -> output projection.
// Inputs: Xq Xk Xv Wq bq Wk bk Wv bv Wh bh ; out = [H | A] flat.
// ---------------------------------------------------------------------------
extern "C" void kernel_launch(void* const* d_in, const int* in_sizes, int n_in,
                              void* d_out, int out_size, void* d_ws, size_t ws_size,
                              hipStream_t stream) {
    (void)in_sizes; (void)n_in; (void)out_size; (void)ws_size;

    const float* Xq = (const float*)d_in[0];
    const float* Xk = (const float*)d_in[1];
    const float* Xv = (const float*)d_in[2];
    const float* Wq = (const float*)d_in[3];
    const float* bq = (const float*)d_in[4];
    const float* Wk = (const float*)d_in[5];
    const float* bk = (const float*)d_in[6];
    const float* Wv = (const float*)d_in[7];
    const float* bv = (const float*)d_in[8];
    const float* Wh = (const float*)d_in[9];
    const float* bh = (const float*)d_in[10];

    float* ws   = (float*)d_ws;           // needs ~25 MB
    float* Qws  = ws;
    float* Kws  = Qws + QKV_ELEMS;
    float* Vws  = Kws + QKV_ELEMS;
    float* Hcat = Vws + QKV_ELEMS;

    float* Hout = (float*)d_out;          // [32,1024,48]
    float* Aout = Hout + H_ELEMS;         // [32,3,1024,1024]

    const dim3 blk(32);                   // one wave32 per workgroup
    const dim3 pgrid(NROWS / 16);         // 2048 row tiles

    ca_proj_kernel<<<pgrid, blk, 0, stream>>>(Xq, Wq, bq, Qws, 1);
    ca_proj_kernel<<<pgrid, blk, 0, stream>>>(Xk, Wk, bk, Kws, 1);
    ca_proj_kernel<<<pgrid, blk, 0, stream>>>(Xv, Wv, bv, Vws, 1);

    ca_attn_kernel<<<dim3(SLEN / 16, NHEADS, BATCH), blk, 0, stream>>>(
        Qws, Kws, Vws, Hcat, Aout);

    ca_proj_kernel<<<pgrid, blk, 0, stream>>>(Hcat, Wh, bh, Hout, 0);
}